// KNNLayer_67164698575279
// MI455X (gfx1250) — compile-verified
//
#include <hip/hip_runtime.h>
#include <hip/hip_bf16.h>
#include <math.h>

// ---------------------------------------------------------------------------
// KNN local-attention layer for MI455X (gfx1250, wave32, WMMA f32 16x16x4)
// Shapes: B=4, C=96, H=W=48, HW=2304, P=9216, window 13x13 (169), K=8, D=192.
// ---------------------------------------------------------------------------

typedef __attribute__((ext_vector_type(2))) float v2f;
typedef __attribute__((ext_vector_type(8))) float v8f;

#define CB   4
#define CC   96
#define CH   48
#define CW   48
#define CHW  (CH * CW)          // 2304
#define CP   (CB * CHW)         // 9216
#define CD   192
#define HALFW 6                 // window half-size (13x13)
#define NCAND 364               // 13 rows x 28 cols union window for 16 pixels
#define NTILE 23                // ceil(364/16)
#define KTOP 8

static __device__ __forceinline__ v8f wmma_f32_k4(v2f a, v2f b, v8f c) {
    // D = A(16x4) * B(4x16) + C, fp32 exact
    return __builtin_amdgcn_wmma_f32_16x16x4_f32(false, a, false, b,
                                                 (short)0, c, false, false);
}

// insert (v,g) into sorted-descending top-8 register lists
static __device__ __forceinline__ void insert8(float v, int g,
                                               float (&tv)[KTOP], int (&tg)[KTOP]) {
    if (v > tv[KTOP - 1]) {
        tv[KTOP - 1] = v; tg[KTOP - 1] = g;
#pragma unroll
        for (int j = KTOP - 1; j > 0; --j) {
            if (tv[j] > tv[j - 1]) {
                float ft = tv[j]; tv[j] = tv[j - 1]; tv[j - 1] = ft;
                int   it = tg[j]; tg[j] = tg[j - 1]; tg[j - 1] = it;
            }
        }
    }
}

// ---------------------------------------------------------------------------
// Kernel 1: per-pixel L2-normalize (x_n) and LayerNorm (x_ln).
// One wave32 per pixel; lane handles channels lane, lane+32, lane+64.
// ---------------------------------------------------------------------------
__global__ __launch_bounds__(128)
void k_prep(const float* __restrict__ x, const float* __restrict__ ln_w,
            const float* __restrict__ ln_b, float* __restrict__ xn,
            float* __restrict__ xln) {
    const int lane = threadIdx.x & 31;
    const int wave = threadIdx.x >> 5;
    const int p = blockIdx.x * 4 + wave;          // 0..9215
    const int b = p / CHW;
    const int hw = p - b * CHW;
    const float* xb = x + (size_t)b * CC * CHW + hw;

    float v0 = xb[(size_t)(lane +  0) * CHW];
    float v1 = xb[(size_t)(lane + 32) * CHW];
    float v2 = xb[(size_t)(lane + 64) * CHW];

    float s  = v0 + v1 + v2;
    float ss = v0 * v0 + v1 * v1 + v2 * v2;
#pragma unroll
    for (int m = 16; m >= 1; m >>= 1) {
        s  += __shfl_xor(s,  m, 32);
        ss += __shfl_xor(ss, m, 32);
    }
    const float inv_n  = 1.0f / fmaxf(sqrtf(ss), 1e-12f);
    const float mean   = s * (1.0f / 96.0f);
    const float var    = fmaxf(ss * (1.0f / 96.0f) - mean * mean, 0.0f);
    const float inv_sd = rsqrtf(var + 1e-5f);

    float* xnp = xn  + (size_t)p * CC;
    float* xlp = xln + (size_t)p * CC;
    xnp[lane +  0] = v0 * inv_n;
    xnp[lane + 32] = v1 * inv_n;
    xnp[lane + 64] = v2 * inv_n;
    xlp[lane +  0] = (v0 - mean) * inv_sd * ln_w[lane +  0] + ln_b[lane +  0];
    xlp[lane + 32] = (v1 - mean) * inv_sd * ln_w[lane + 32] + ln_b[lane + 32];
    xlp[lane + 64] = (v2 - mean) * inv_sd * ln_w[lane + 64] + ln_b[lane + 64];
}

// ---------------------------------------------------------------------------
// Kernel 2 helpers: candidate-tile A-operand load, and the WMMA + top-8 step.
// ---------------------------------------------------------------------------
static __device__ __forceinline__ void sims_load_tile(
        const float* __restrict__ xn, int base, int py, int x0,
        int half, int l16, int t, v2f (&aop)[24]) {
    const int cA  = t * 16 + l16;
    const int wyA = cA / 28;
    const int wxA = cA - wyA * 28;
    const int cy  = min(max(py - HALFW + wyA, 0), CH - 1);
    const int cx  = min(max(x0 - HALFW + wxA, 0), CW - 1);
    const float* arow = xn + (size_t)(base + cy * CW + cx) * CC + 2 * half;
#pragma unroll
    for (int kk = 0; kk < 24; ++kk)
        aop[kk] = *(const v2f*)(arow + 4 * kk);
}

static __device__ __forceinline__ void sims_process_tile(
        const v2f (&aop)[24], const v2f (&bop)[24], int t, int py, int x0,
        int half, int l16, float (&tv)[KTOP], int (&tg)[KTOP]) {
    // 4 independent accumulator chains -> XDL pipelining
    v8f a0 = {}, a1 = {}, a2 = {}, a3 = {};
#pragma unroll
    for (int kk = 0; kk < 24; kk += 4) {
        a0 = wmma_f32_k4(aop[kk + 0], bop[kk + 0], a0);
        a1 = wmma_f32_k4(aop[kk + 1], bop[kk + 1], a1);
        a2 = wmma_f32_k4(aop[kk + 2], bop[kk + 2], a2);
        a3 = wmma_f32_k4(aop[kk + 3], bop[kk + 3], a3);
    }
    const v8f acc = (a0 + a1) + (a2 + a3);

    // D[M=cand][N=pixel]: lane holds rows m = i + 8*half for pixel l16
#pragma unroll
    for (int i = 0; i < 8; ++i) {
        const int cm = t * 16 + i + 8 * half;
        if (cm < NCAND) {
            const int wy = cm / 28;
            const int wx = cm - wy * 28;
            const int ny = py - HALFW + wy;
            const int nx = x0 - HALFW + wx;
            const int dxp = wx - HALFW - l16;   // nx - px of this pixel
            const bool ok = (ny >= 0) & (ny < CH) & (nx >= 0) & (nx < CW) &
                            (dxp >= -HALFW) & (dxp <= HALFW);
            if (ok) insert8(acc[i], ny * CW + nx, tv, tg);
        }
    }
}

// ---------------------------------------------------------------------------
// Kernel 2: WMMA cosine-sim scores over the union window, per-pixel top-8,
// softmax, weighted feature aggregation, + x_ln residual -> enhanced.
// One wave32 per group of 16 consecutive pixels along x. 576 groups.
// Candidate tiles are software-pipelined (double-buffered A operand).
// ---------------------------------------------------------------------------
__global__ __launch_bounds__(32)
void k_sims(const float* __restrict__ x, const float* __restrict__ xn,
            const float* __restrict__ xln, float* __restrict__ enh) {
    __shared__ int   s_idx[16][KTOP];
    __shared__ float s_wgt[16][KTOP];

    const int g   = blockIdx.x;            // 0..575
    const int gx  = g % 3;
    const int py  = (g / 3) % CH;
    const int b   = g / (3 * CH);
    const int x0  = gx * 16;
    const int lane = threadIdx.x;
    const int half = lane >> 4;            // 0 | 1 -> K-pair {0,1} | {2,3}
    const int l16  = lane & 15;
    const int base = b * CHW;
    const int p0   = base + py * CW + x0;  // first pixel of group (global flat)

    // B operand: this lane's pixel row of x_n (resident across all tiles)
    v2f bop[24];
    {
        const float* prow = xn + (size_t)(p0 + l16) * CC + 2 * half;
#pragma unroll
        for (int kk = 0; kk < 24; ++kk)
            bop[kk] = *(const v2f*)(prow + 4 * kk);
    }

    float tv[KTOP]; int tg[KTOP];
#pragma unroll
    for (int j = 0; j < KTOP; ++j) { tv[j] = -3.0e38f; tg[j] = 0; }

    // software pipeline over the 23 candidate tiles (ping-pong A buffers)
    v2f aopA[24], aopB[24];
    sims_load_tile(xn, base, py, x0, half, l16, 0, aopA);
    int t = 0;
    while (true) {
        if (t + 1 < NTILE) sims_load_tile(xn, base, py, x0, half, l16, t + 1, aopB);
        sims_process_tile(aopA, bop, t, py, x0, half, l16, tv, tg);
        ++t;
        if (t >= NTILE) break;
        if (t + 1 < NTILE) sims_load_tile(xn, base, py, x0, half, l16, t + 1, aopA);
        sims_process_tile(aopB, bop, t, py, x0, half, l16, tv, tg);
        ++t;
        if (t >= NTILE) break;
    }

    // merge the two half-wave candidate streams for the same pixel
#pragma unroll
    for (int j = 0; j < KTOP; ++j) {
        const float pv = __shfl_xor(tv[j], 16, 32);
        const int   pg = __shfl_xor(tg[j], 16, 32);
        insert8(pv, pg, tv, tg);
    }

    // softmax over sorted-descending top-8 (tv[0] is max; >=49 valid cands)
    float w[KTOP]; float wsum = 0.0f;
#pragma unroll
    for (int j = 0; j < KTOP; ++j) { w[j] = __expf(tv[j] - tv[0]); wsum += w[j]; }
    const float winv = 1.0f / wsum;
    if (half == 0) {
#pragma unroll
        for (int j = 0; j < KTOP; ++j) { s_idx[l16][j] = tg[j]; s_wgt[l16][j] = w[j] * winv; }
    }
    __syncthreads();

    // aggregation: each lane handles 3 channels of each of the 16 pixels
    const int c0 = lane * 3;
    const float* xb = x + (size_t)b * CC * CHW;
    for (int n = 0; n < 16; ++n) {
        const int pp = p0 + n;
        float a0 = 0.f, a1 = 0.f, a2 = 0.f;
#pragma unroll
        for (int j = 0; j < KTOP; ++j) {
            const int   gg = s_idx[n][j];
            const float ww = s_wgt[n][j];
            a0 += ww * xb[(size_t)(c0 + 0) * CHW + gg];
            a1 += ww * xb[(size_t)(c0 + 1) * CHW + gg];
            a2 += ww * xb[(size_t)(c0 + 2) * CHW + gg];
        }
        const float* xl = xln + (size_t)pp * CC + c0;
        float*       eo = enh + (size_t)pp * CC + c0;
        eo[0] = a0 + xl[0];
        eo[1] = a1 + xl[1];
        eo[2] = a2 + xl[2];
    }
}

// ---------------------------------------------------------------------------
// Kernel 3: h = relu(enhanced(9216x96) @ w1^T(96x192) + b1).
// One wave per 16x16 output tile, K=96 over 24 WMMAs (4 chains).
// ---------------------------------------------------------------------------
__global__ __launch_bounds__(256)
void k_ffn1(const float* __restrict__ enh, const float* __restrict__ w1,
            const float* __restrict__ b1, float* __restrict__ h) {
    const int wid  = (blockIdx.x * blockDim.x + threadIdx.x) >> 5;   // 0..6911
    const int lane = threadIdx.x & 31;
    const int half = lane >> 4;
    const int l16  = lane & 15;
    const int td   = wid % (CD / 16);
    const int tp   = wid / (CD / 16);
    const int p0   = tp * 16;
    const int d0   = td * 16;

    const float* arow = enh + (size_t)(p0 + l16) * CC + 2 * half;   // A: pixels
    const float* brow = w1  + (size_t)(d0 + l16) * CC + 2 * half;   // B: outputs

    v8f a0 = {}, a1 = {}, a2 = {}, a3 = {};
#pragma unroll
    for (int kk = 0; kk < 24; kk += 4) {
        a0 = wmma_f32_k4(*(const v2f*)(arow + 4 * (kk + 0)),
                         *(const v2f*)(brow + 4 * (kk + 0)), a0);
        a1 = wmma_f32_k4(*(const v2f*)(arow + 4 * (kk + 1)),
                         *(const v2f*)(brow + 4 * (kk + 1)), a1);
        a2 = wmma_f32_k4(*(const v2f*)(arow + 4 * (kk + 2)),
                         *(const v2f*)(brow + 4 * (kk + 2)), a2);
        a3 = wmma_f32_k4(*(const v2f*)(arow + 4 * (kk + 3)),
                         *(const v2f*)(brow + 4 * (kk + 3)), a3);
    }
    const v8f acc = (a0 + a1) + (a2 + a3);
    const float bias = b1[d0 + l16];
#pragma unroll
    for (int i = 0; i < 8; ++i) {
        const int m = i + 8 * half;
        h[(size_t)(p0 + m) * CD + d0 + l16] = fmaxf(acc[i] + bias, 0.0f);
    }
}

// ---------------------------------------------------------------------------
// Kernel 4: out = enhanced + h(9216x192) @ w2^T(192x96) + b2,
// stored transposed back to (B, C, H, W). K=192 -> 48 WMMAs (4 chains).
// ---------------------------------------------------------------------------
__global__ __launch_bounds__(256)
void k_ffn2(const float* __restrict__ h, const float* __restrict__ w2,
            const float* __restrict__ b2, const float* __restrict__ enh,
            float* __restrict__ out) {
    const int wid  = (blockIdx.x * blockDim.x + threadIdx.x) >> 5;   // 0..3455
    const int lane = threadIdx.x & 31;
    const int half = lane >> 4;
    const int l16  = lane & 15;
    const int tc   = wid % (CC / 16);
    const int tp   = wid / (CC / 16);
    const int p0   = tp * 16;
    const int c0   = tc * 16;

    const float* arow = h  + (size_t)(p0 + l16) * CD + 2 * half;    // A: pixels
    const float* brow = w2 + (size_t)(c0 + l16) * CD + 2 * half;    // B: channels

    v8f a0 = {}, a1 = {}, a2 = {}, a3 = {};
#pragma unroll
    for (int kk = 0; kk < 48; kk += 4) {
        a0 = wmma_f32_k4(*(const v2f*)(arow + 4 * (kk + 0)),
                         *(const v2f*)(brow + 4 * (kk + 0)), a0);
        a1 = wmma_f32_k4(*(const v2f*)(arow + 4 * (kk + 1)),
                         *(const v2f*)(brow + 4 * (kk + 1)), a1);
        a2 = wmma_f32_k4(*(const v2f*)(arow + 4 * (kk + 2)),
                         *(const v2f*)(brow + 4 * (kk + 2)), a2);
        a3 = wmma_f32_k4(*(const v2f*)(arow + 4 * (kk + 3)),
                         *(const v2f*)(brow + 4 * (kk + 3)), a3);
    }
    const v8f acc = (a0 + a1) + (a2 + a3);
    const float bias = b2[c0 + l16];
    const int b   = p0 / CHW;
    const int hw0 = p0 - b * CHW;            // 16 | CHW so tile stays in one b
    float* ob = out + (size_t)b * CC * CHW + (size_t)(c0 + l16) * CHW;
#pragma unroll
    for (int i = 0; i < 8; ++i) {
        const int m = i + 8 * half;
        const float e = enh[(size_t)(p0 + m) * CC + c0 + l16];
        ob[hw0 + m] = acc[i] + bias + e;
    }
}

// ---------------------------------------------------------------------------
extern "C" void kernel_launch(void* const* d_in, const int* in_sizes, int n_in,
                              void* d_out, int out_size, void* d_ws, size_t ws_size,
                              hipStream_t stream) {
    const float* x    = (const float*)d_in[0];
    const float* w1   = (const float*)d_in[1];
    const float* b1   = (const float*)d_in[2];
    const float* w2   = (const float*)d_in[3];
    const float* b2   = (const float*)d_in[4];
    const float* ln_w = (const float*)d_in[5];
    const float* ln_b = (const float*)d_in[6];
    float* out = (float*)d_out;

    float* ws  = (float*)d_ws;
    float* xn  = ws;                                   // 9216*96
    float* xln = xn  + (size_t)CP * CC;                // 9216*96
    float* enh = xln + (size_t)CP * CC;                // 9216*96
    float* hbf = enh + (size_t)CP * CC;                // 9216*192

    k_prep<<<CP / 4, 128, 0, stream>>>(x, ln_w, ln_b, xn, xln);
    k_sims<<<CB * CH * (CW / 16), 32, 0, stream>>>(x, xn, xln, enh);
    k_ffn1<<<(CP / 16) * (CD / 16) / 8, 256, 0, stream>>>(enh, w1, b1, hbf);
    k_ffn2<<<(CP / 16) * (CC / 16) / 8, 256, 0, stream>>>(hbf, w2, b2, enh, out);
}